// KernalAnsatz_65481071403018
// MI455X (gfx1250) — compile-verified
//
#include <hip/hip_runtime.h>
#include <cstdint>
#include <cstddef>

// RBF kernel matrix: K[i,j] = exp(-gamma * ||x_i - y_j||^2)
// x: [8192, 512] f32, y: [8192, 512] f32, out: [8192, 8192] f32.
// Store-bound (~256MB out @ 23.3 TB/s ~= 12us); GEMM part done in full f32
// precision with V_WMMA_F32_16X16X4_F32.

typedef __attribute__((ext_vector_type(2))) float v2f;
typedef __attribute__((ext_vector_type(8))) float v8f;

#define D_DIM 512
#define BM 128            // block tile M
#define BN 64             // block tile N
#define KC 32             // K chunk staged in LDS
#define LDX 36            // padded LDS row stride (floats): 16B aligned, bank-conflict free
#define LDY 36

// ---------------- Kernel 1: per-row sum of squares for x and y ----------------
__global__ __launch_bounds__(256) void rbf_rowsumsq(const float* __restrict__ x,
                                                    const float* __restrict__ y,
                                                    float* __restrict__ sums,
                                                    int nx, int ny) {
  int row  = blockIdx.x * 8 + (threadIdx.x >> 5);   // one wave32 per row
  int lane = threadIdx.x & 31;
  if (row >= nx + ny) return;
  const float* src = (row < nx) ? x : y;
  int r = (row < nx) ? row : row - nx;
  const float4* p = reinterpret_cast<const float4*>(src + (size_t)r * D_DIM);
  float s = 0.0f;
#pragma unroll
  for (int i = 0; i < (D_DIM / 4) / 32; ++i) {      // 4 float4 per lane
    float4 v = p[lane + i * 32];
    s += v.x * v.x + v.y * v.y + v.z * v.z + v.w * v.w;
  }
#pragma unroll
  for (int off = 16; off > 0; off >>= 1) s += __shfl_xor(s, off, 32);
  if (lane == 0) sums[row] = s;
}

// ---------------- Kernel 2: WMMA f32 GEMM + RBF epilogue ----------------
__global__ __launch_bounds__(256) void rbf_wmma_gemm(const float* __restrict__ x,
                                                     const float* __restrict__ y,
                                                     const float* __restrict__ x2,
                                                     const float* __restrict__ y2,
                                                     const float* __restrict__ gptr,
                                                     float* __restrict__ out,
                                                     int ny) {
  __shared__ float lx[BM * LDX];
  __shared__ float ly[BN * LDY];

  const int tid  = threadIdx.x;
  const int lane = tid & 31;
  const int wave = tid >> 5;     // 8 waves
  const int wm   = wave & 3;     // 0..3 -> 32-row slab of the 128-row block tile
  const int wn   = wave >> 2;    // 0..1 -> 32-col slab of the 64-col block tile
  const int lh   = lane >> 4;    // half-wave id (0/1)
  const int l15  = lane & 15;

  const int m_base = blockIdx.y * BM;
  const int n_base = blockIdx.x * BN;

  const float gamma = *gptr;

  v8f c00 = {}; v8f c01 = {}; v8f c10 = {}; v8f c11 = {};

  // A frag (16x4 f32): lanes 0-15 row=M, v0/v1 = K0/K1; lanes 16-31 = K2/K3.
  // B frag (4x16 f32): symmetric over columns. Both read an aligned 8B pair.
  const int am0  = wm * 32 + l15;      // row within block tile (tile1 adds +16)
  const int bn0  = wn * 32 + l15;      // col within block tile
  const int koff = lh * 2;

  for (int k0 = 0; k0 < D_DIM; k0 += KC) {
    __syncthreads();
    // Stage x block: BM x KC floats, float4 granularity, coalesced.
    for (int i = tid; i < BM * (KC / 4); i += 256) {
      int r = i >> 3, c4 = i & 7;
      float4 v = *reinterpret_cast<const float4*>(
          x + (size_t)(m_base + r) * D_DIM + k0 + c4 * 4);
      *reinterpret_cast<float4*>(&lx[r * LDX + c4 * 4]) = v;
    }
    // Stage y block: BN x KC floats.
    for (int i = tid; i < BN * (KC / 4); i += 256) {
      int r = i >> 3, c4 = i & 7;
      float4 v = *reinterpret_cast<const float4*>(
          y + (size_t)(n_base + r) * D_DIM + k0 + c4 * 4);
      *reinterpret_cast<float4*>(&ly[r * LDY + c4 * 4]) = v;
    }
    __syncthreads();

#pragma unroll
    for (int kk = 0; kk < KC; kk += 4) {
      v2f a0 = *reinterpret_cast<const v2f*>(&lx[(am0)      * LDX + kk + koff]);
      v2f a1 = *reinterpret_cast<const v2f*>(&lx[(am0 + 16) * LDX + kk + koff]);
      v2f b0 = *reinterpret_cast<const v2f*>(&ly[(bn0)      * LDY + kk + koff]);
      v2f b1 = *reinterpret_cast<const v2f*>(&ly[(bn0 + 16) * LDY + kk + koff]);
      c00 = __builtin_amdgcn_wmma_f32_16x16x4_f32(false, a0, false, b0,
                                                  (short)0, c00, false, false);
      c01 = __builtin_amdgcn_wmma_f32_16x16x4_f32(false, a0, false, b1,
                                                  (short)0, c01, false, false);
      c10 = __builtin_amdgcn_wmma_f32_16x16x4_f32(false, a1, false, b0,
                                                  (short)0, c10, false, false);
      c11 = __builtin_amdgcn_wmma_f32_16x16x4_f32(false, a1, false, b1,
                                                  (short)0, c11, false, false);
    }
  }

  // Epilogue: sq = max(x2 + y2 - 2*dot, 0); out = exp(-gamma*sq).
  // C/D layout: VGPR r -> M = r + 8*lh, N = l15 (per 16x16 tile).
  const float ng = -gamma;
  auto emit = [&](const v8f& c, int mt, int nt) {
    int col = n_base + wn * 32 + nt * 16 + l15;
    float y2v = y2[col];
#pragma unroll
    for (int r = 0; r < 8; ++r) {
      int row = m_base + wm * 32 + mt * 16 + lh * 8 + r;
      float sq = x2[row] + y2v - 2.0f * c[r];
      sq = fmaxf(sq, 0.0f);
      float res = __expf(ng * sq);
      __builtin_nontemporal_store(res, out + (size_t)row * ny + col);
    }
  };
  emit(c00, 0, 0); emit(c01, 0, 1); emit(c10, 1, 0); emit(c11, 1, 1);
}

// ---------------- Host launch ----------------
extern "C" void kernel_launch(void* const* d_in, const int* in_sizes, int n_in,
                              void* d_out, int out_size, void* d_ws, size_t ws_size,
                              hipStream_t stream) {
  const float* x = (const float*)d_in[0];
  const float* y = (const float*)d_in[1];
  const float* g = (const float*)d_in[2];   // 1-element gamma
  float* out = (float*)d_out;

  const int nx = in_sizes[0] / D_DIM;       // 8192
  const int ny = in_sizes[1] / D_DIM;       // 8192

  float* sums = (float*)d_ws;               // nx + ny floats (64 KB)
  const int rows = nx + ny;
  rbf_rowsumsq<<<(rows + 7) / 8, 256, 0, stream>>>(x, y, sums, nx, ny);

  dim3 grid(ny / BN, nx / BM);              // (128, 64)
  rbf_wmma_gemm<<<grid, 256, 0, stream>>>(x, y, sums, sums + nx, g, out, ny);
}